// Experts_44401371906336
// MI455X (gfx1250) — compile-verified
//
#include <hip/hip_runtime.h>
#include <hip/hip_bf16.h>

// Problem constants (from reference)
constexpr int E_   = 8;
constexpr int N_   = 4096;
constexpr int D_   = 1024;
constexpr int DFF_ = 2048;
constexpr int NE_  = 65536;

typedef __attribute__((ext_vector_type(2))) float v2f;
typedef __attribute__((ext_vector_type(8))) float v8f;

// ---------------- GEMM: C[e] = (relu?)(A[e] @ B[e]) ----------------
// A: M x K row-major, B: K x Nd row-major, C: M x Nd row-major.
// Block tile 128x64, K-tile 16, double-buffered LDS. 256 threads = 8 wave32.
// Wave grid 4(m) x 2(n); each wave computes 32x32 via 2x2 wmma f32 16x16x4 tiles.

#define BM 128
#define BN 64
#define BK 16
#define LDS_STRIDE 20   // pad: bank-conflict-free b64 fragment loads, 16B-aligned f4 stores

__global__ __launch_bounds__(256)
void gemm_f32_wmma_kernel(const float* __restrict__ A,
                          const float* __restrict__ B,
                          float* __restrict__ C,
                          int M, int Nd, int K,
                          long long strideA, long long strideB, long long strideC,
                          int do_relu)
{
    __shared__ float lA[2][BM * LDS_STRIDE];   // 2 * 10240 B
    __shared__ float lB[2][BN * LDS_STRIDE];   // 2 *  5120 B (transposed: [n][k])

    const int e = blockIdx.z;
    A += (long long)e * strideA;
    B += (long long)e * strideB;
    C += (long long)e * strideC;

    const int tid   = threadIdx.x;
    const int lane  = tid & 31;
    const int wave  = tid >> 5;
    const int wm    = (wave & 3) << 5;      // 0,32,64,96
    const int wn    = (wave >> 2) << 5;     // 0,32
    const int lhalf = lane & 15;            // row/col within 16-wide tile
    const int khalf = (lane >> 4) << 1;     // 0 or 2 (K sub-offset per ISA A/B layout)

    const int m0 = blockIdx.y * BM;
    const int n0 = blockIdx.x * BN;

    // per-thread staging coordinates (fixed across K-chunks)
    const int ar0 = tid >> 2;               // A rows 0..63
    const int ar1 = (tid + 256) >> 2;       // A rows 64..127
    const int ac4 = (tid & 3) << 2;         // A col 0,4,8,12
    const int br  = tid >> 4;               // B k-row 0..15
    const int bc4 = (tid & 15) << 2;        // B n-col 0..60

    const float* gA0 = A + (long long)(m0 + ar0) * K + ac4;
    const float* gA1 = A + (long long)(m0 + ar1) * K + ac4;
    const float* gB  = B + (long long)br * Nd + n0 + bc4;
    const long long gBstep = (long long)BK * Nd;

    v8f acc[2][2] = {};

    // preload chunk 0 into buffer 0
    float4 va0 = *(const float4*)(gA0);
    float4 va1 = *(const float4*)(gA1);
    float4 vb  = *(const float4*)(gB);
    {
        *(float4*)(&lA[0][ar0 * LDS_STRIDE + ac4]) = va0;
        *(float4*)(&lA[0][ar1 * LDS_STRIDE + ac4]) = va1;
        lB[0][(bc4 + 0) * LDS_STRIDE + br] = vb.x;
        lB[0][(bc4 + 1) * LDS_STRIDE + br] = vb.y;
        lB[0][(bc4 + 2) * LDS_STRIDE + br] = vb.z;
        lB[0][(bc4 + 3) * LDS_STRIDE + br] = vb.w;
    }
    __syncthreads();

    int cur = 0;
    for (int k0 = 0; k0 < K; k0 += BK) {
        const bool has_next = (k0 + BK) < K;
        // prefetch next chunk into registers (overlaps with WMMA issue)
        if (has_next) {
            va0 = *(const float4*)(gA0 + k0 + BK);
            va1 = *(const float4*)(gA1 + k0 + BK);
            vb  = *(const float4*)(gB + ((long long)(k0 / BK) + 1) * gBstep);
        }

        // compute on current buffer
        const float* cA = lA[cur];
        const float* cB = lB[cur];
#pragma unroll
        for (int kk = 0; kk < BK; kk += 4) {
            v2f a[2], b[2];
#pragma unroll
            for (int i = 0; i < 2; ++i)
                a[i] = *(const v2f*)(&cA[(wm + (i << 4) + lhalf) * LDS_STRIDE + kk + khalf]);
#pragma unroll
            for (int j = 0; j < 2; ++j)
                b[j] = *(const v2f*)(&cB[(wn + (j << 4) + lhalf) * LDS_STRIDE + kk + khalf]);
#pragma unroll
            for (int i = 0; i < 2; ++i)
#pragma unroll
                for (int j = 0; j < 2; ++j)
                    acc[i][j] = __builtin_amdgcn_wmma_f32_16x16x4_f32(
                        false, a[i], false, b[j], (short)0, acc[i][j], false, false);
        }

        // stage next chunk into the other buffer
        if (has_next) {
            const int nxt = cur ^ 1;
            *(float4*)(&lA[nxt][ar0 * LDS_STRIDE + ac4]) = va0;
            *(float4*)(&lA[nxt][ar1 * LDS_STRIDE + ac4]) = va1;
            lB[nxt][(bc4 + 0) * LDS_STRIDE + br] = vb.x;
            lB[nxt][(bc4 + 1) * LDS_STRIDE + br] = vb.y;
            lB[nxt][(bc4 + 2) * LDS_STRIDE + br] = vb.z;
            lB[nxt][(bc4 + 3) * LDS_STRIDE + br] = vb.w;
            __syncthreads();
            cur = nxt;
        }
    }

    // --- writeback per ISA C/D layout: VGPR r -> row r (lanes 0-15) / row 8+r (lanes 16-31)
    const int rbase = (lane >> 4) << 3;   // 0 or 8
#pragma unroll
    for (int i = 0; i < 2; ++i) {
#pragma unroll
        for (int j = 0; j < 2; ++j) {
            const int row0 = m0 + wm + (i << 4) + rbase;
            const int col  = n0 + wn + (j << 4) + lhalf;
#pragma unroll
            for (int r = 0; r < 8; ++r) {
                float v = acc[i][j][r];
                if (do_relu) v = fmaxf(v, 0.0f);   // v_max_num_f32
                C[(long long)(row0 + r) * Nd + col] = v;
            }
        }
    }
}

// ---------------- edge scatter: out[e, dst, :] += y[e, src, :] ----------------
// rows are D_=1024 floats: exactly one float4 per thread (256 threads)
__global__ __launch_bounds__(256)
void scatter_add_kernel(const float* __restrict__ y,
                        const long long* __restrict__ edges,
                        float* __restrict__ out)
{
    const int edge = blockIdx.x;
    const int e    = blockIdx.y;
    const long long src = edges[edge];
    const long long dst = edges[NE_ + edge];

    const float* ys = y   + ((long long)e * N_ + src) * D_;
    float*       od = out + ((long long)e * N_ + dst) * D_;

    const int c = threadIdx.x * 4;
    float4 v = *(const float4*)(ys + c);
    atomicAdd(od + c + 0, v.x);
    atomicAdd(od + c + 1, v.y);
    atomicAdd(od + c + 2, v.z);
    atomicAdd(od + c + 3, v.w);
}

extern "C" void kernel_launch(void* const* d_in, const int* in_sizes, int n_in,
                              void* d_out, int out_size, void* d_ws, size_t ws_size,
                              hipStream_t stream)
{
    const float*     x     = (const float*)d_in[0];      // [E, N, D]
    const float*     W1    = (const float*)d_in[1];      // [E, D, DFF]
    const float*     W2    = (const float*)d_in[2];      // [E, DFF, D]
    const long long* edges = (const long long*)d_in[3];  // [2, NE]
    float*           out   = (float*)d_out;              // [E, N, D]

    float* h = (float*)d_ws;                             // [E, N, DFF]  268 MB
    float* y = h + (size_t)E_ * N_ * DFF_;               // [E, N, D]    134 MB

    // Phase 1: h = relu(x @ W1)   (M=N_, Nd=DFF_, K=D_)
    {
        dim3 grid(DFF_ / BN, N_ / BM, E_);   // (32, 32, 8)
        gemm_f32_wmma_kernel<<<grid, 256, 0, stream>>>(
            x, W1, h, N_, DFF_, D_,
            (long long)N_ * D_, (long long)D_ * DFF_, (long long)N_ * DFF_, 1);
    }

    // Phase 2: y = h @ W2   (M=N_, Nd=D_, K=DFF_)  -- associativity: out = S·(h·W2)
    {
        dim3 grid(D_ / BN, N_ / BM, E_);     // (16, 32, 8)
        gemm_f32_wmma_kernel<<<grid, 256, 0, stream>>>(
            h, W2, y, N_, D_, DFF_,
            (long long)N_ * DFF_, (long long)DFF_ * D_, (long long)N_ * D_, 0);
    }

    // Phase 3: out = segment_sum(y[src] -> dst)  (rows of D_ floats)
    hipMemsetAsync(out, 0, (size_t)E_ * N_ * D_ * sizeof(float), stream);
    {
        dim3 grid(NE_, E_);
        scatter_add_kernel<<<grid, 256, 0, stream>>>(y, edges, out);
    }
}